// JointNetwork_41403484733770
// MI455X (gfx1250) — compile-verified
//
#include <hip/hip_runtime.h>

// ---------------------------------------------------------------------------
// RNN-T Joint network for MI455X (gfx1250, wave32, WMMA).
//   a' [B*T][64] = acoustic @ W_a + b1
//   p' [B*S][64] = prediction @ W_p
//   out[b,t,s,:] = relu(a'[b,t,:] + p'[b,s,:]) @ W_out + b2
// Joint stage uses V_WMMA_F32_16X16X4_F32 (exact f32), store-bandwidth bound.
// ---------------------------------------------------------------------------

typedef float v2f __attribute__((ext_vector_type(2)));
typedef float v8f __attribute__((ext_vector_type(8)));

#define B_DIM 8
#define T_DIM 512
#define S_DIM 128
#define C_DIM 512
#define J_DIM 64

// ---------------------------------------------------------------------------
// Stage 1: Y[row][j] = sum_c X[row][c] * W[c][j] (+ bias[j])
// M rows of K=512 -> 64 outputs. Tiny (<0.35 GFLOP total), plain FMA.
// Block = 256 threads = 4 rows x 64 cols.
// ---------------------------------------------------------------------------
__global__ __launch_bounds__(256) void proj_kernel(
    const float* __restrict__ X, const float* __restrict__ W,
    const float* __restrict__ bias, float* __restrict__ Y, int M) {
  const int row = blockIdx.x * 4 + (threadIdx.x >> 6);
  const int j = threadIdx.x & 63;
  if (row >= M) return;
  const float* x = X + (size_t)row * C_DIM;
  float acc = bias ? bias[j] : 0.0f;
#pragma unroll 4
  for (int c = 0; c < C_DIM; c += 4) {
    const float4 xv = *(const float4*)(x + c);
    acc += xv.x * W[(c + 0) * J_DIM + j];
    acc += xv.y * W[(c + 1) * J_DIM + j];
    acc += xv.z * W[(c + 2) * J_DIM + j];
    acc += xv.w * W[(c + 3) * J_DIM + j];
  }
  Y[(size_t)row * J_DIM + j] = acc;
}

// ---------------------------------------------------------------------------
// Stage 2: joint GEMM. One block per (b,t). 8 waves x 16 s-rows = S=128.
// Wave tile: M=16 (s), N=64 (4 n-tiles of 16), K=64 (16 steps of 4).
// A-frag (16x4 f32): lane%16 = M, lane/16 selects K{0,1}/{2,3} -> built in
// registers straight from global (relu(a'+p')), no LDS transpose needed.
// B-frag (4x16 f32): lane%16 = N, (lane/16, vgpr) = K -> W_out pre-swizzled
// into LDS in fragment order, each step is one conflict-free b64 LDS read.
// ---------------------------------------------------------------------------
__global__ __launch_bounds__(256) void joint_kernel(
    const float* __restrict__ aprime,  // [B*T][64]  (a@W_a + b1)
    const float* __restrict__ pj,      // [B*S][64]  (p@W_p)
    const float* __restrict__ Wout,    // [64][64]   row-major [k][n]
    const float* __restrict__ b2,      // [64]
    float* __restrict__ out)           // [B*T*S][64]
{
  __shared__ float bfrag[16][4][32][2];  // [kb][nt][lane][v] = 16 KB

  const int tid = threadIdx.x;
  // Swizzle W_out into B-fragment order (4096 floats, 16 per thread).
  for (int i = tid; i < 16 * 4 * 32 * 2; i += 256) {
    const int v = i & 1;
    const int lane = (i >> 1) & 31;
    const int nt = (i >> 6) & 3;
    const int kb = (i >> 8);
    const int k = 4 * kb + 2 * (lane >> 4) + v;
    const int n = 16 * nt + (lane & 15);
    bfrag[kb][nt][lane][v] = Wout[k * J_DIM + n];
  }
  __syncthreads();

  const int bt = blockIdx.x;  // 0 .. B*T-1
  const int b = bt / T_DIM;
  const int lane = tid & 31;
  const int s0 = (tid >> 5) * 16;         // wave -> s-tile
  const int row_s = s0 + (lane & 15);     // this lane's A-row (M = lane%16)
  const int joff = 2 * (lane >> 4);       // K sub-offset 0 or 2

  const float* arow = aprime + (size_t)bt * J_DIM;
  const float* prow = pj + ((size_t)b * S_DIM + row_s) * J_DIM;

  // Build A-fragments in registers: afrag[kb] = relu(a'+p') at j = 4kb+joff+{0,1}
  v2f afrag[16];
#pragma unroll
  for (int kb = 0; kb < 16; ++kb) {
    const int j = 4 * kb + joff;
    const v2f av = *(const v2f*)(arow + j);
    const v2f pv = *(const v2f*)(prow + j);
    v2f h;
    h.x = fmaxf(av.x + pv.x, 0.0f);
    h.y = fmaxf(av.y + pv.y, 0.0f);
    afrag[kb] = h;
  }

  v8f acc0 = {}, acc1 = {}, acc2 = {}, acc3 = {};
#pragma unroll
  for (int kb = 0; kb < 16; ++kb) {
    const v2f b0 = *(const v2f*)&bfrag[kb][0][lane][0];
    const v2f b1f = *(const v2f*)&bfrag[kb][1][lane][0];
    const v2f b2f = *(const v2f*)&bfrag[kb][2][lane][0];
    const v2f b3f = *(const v2f*)&bfrag[kb][3][lane][0];
    acc0 = __builtin_amdgcn_wmma_f32_16x16x4_f32(false, afrag[kb], false, b0,
                                                 (short)0, acc0, false, false);
    acc1 = __builtin_amdgcn_wmma_f32_16x16x4_f32(false, afrag[kb], false, b1f,
                                                 (short)0, acc1, false, false);
    acc2 = __builtin_amdgcn_wmma_f32_16x16x4_f32(false, afrag[kb], false, b2f,
                                                 (short)0, acc2, false, false);
    acc3 = __builtin_amdgcn_wmma_f32_16x16x4_f32(false, afrag[kb], false, b3f,
                                                 (short)0, acc3, false, false);
  }

  // Epilogue: C/D layout -> VGPR v holds M = v + 8*(lane>=16), N = lane%16.
  const int m_hi = (lane >> 4) * 8;
  const int n_lo = lane & 15;
  float* obase = out + (size_t)bt * S_DIM * J_DIM;
  const v8f accs[4] = {acc0, acc1, acc2, acc3};
#pragma unroll
  for (int nt = 0; nt < 4; ++nt) {
    const int col = 16 * nt + n_lo;
    const float bias = b2[col];
    const v8f acc = accs[nt];
#pragma unroll
    for (int v = 0; v < 8; ++v) {
      const int s = s0 + m_hi + v;
      __builtin_nontemporal_store(acc[v] + bias,
                                  obase + (size_t)s * J_DIM + col);
    }
  }
}

extern "C" void kernel_launch(void* const* d_in, const int* in_sizes, int n_in,
                              void* d_out, int out_size, void* d_ws,
                              size_t ws_size, hipStream_t stream) {
  const float* acoustic = (const float*)d_in[0];    // [8,512,512]
  const float* prediction = (const float*)d_in[1];  // [8,128,512]
  const float* W_a = (const float*)d_in[2];         // [512,64]
  const float* W_p = (const float*)d_in[3];         // [512,64]
  const float* b1 = (const float*)d_in[4];          // [64]
  const float* W_out = (const float*)d_in[5];       // [64,64]
  const float* b2 = (const float*)d_in[6];          // [64]
  float* out = (float*)d_out;                       // [8,512,128,64]

  float* aprime = (float*)d_ws;                       // 4096*64 floats (1 MB)
  float* pj = aprime + (size_t)B_DIM * T_DIM * J_DIM; // 1024*64 floats (256 KB)

  const int MA = B_DIM * T_DIM;  // 4096
  const int MP = B_DIM * S_DIM;  // 1024

  proj_kernel<<<MA / 4, 256, 0, stream>>>(acoustic, W_a, b1, aprime, MA);
  proj_kernel<<<MP / 4, 256, 0, stream>>>(prediction, W_p, nullptr, pj, MP);
  joint_kernel<<<MA, 256, 0, stream>>>(aprime, pj, W_out, b2, out);
}